// Encoder_22376779612179
// MI455X (gfx1250) — compile-verified
//
#include <hip/hip_runtime.h>
#include <hip/hip_bf16.h>

// Problem constants (B=2, S=2048, D=1024, H=16, AD=64)
#define SEQ   2048
#define DMODEL 1024
#define NHEAD 16
#define ADIM  64
#define BH    32            // B * H
#define NROWS 4096          // B * S
#define ZSIZE (4194304)     // B*H*S*AD floats

typedef __attribute__((ext_vector_type(2))) float v2f;
typedef __attribute__((ext_vector_type(8))) float v8f;

__device__ __forceinline__ v8f wmma_f32_k4(v2f a, v2f b, v8f c) {
  // D = A(16x4) x B(4x16) + C(16x16), f32 WMMA (CDNA5)
  return __builtin_amdgcn_wmma_f32_16x16x4_f32(
      /*neg_a=*/false, a, /*neg_b=*/false, b,
      /*c_mod=*/(short)0, c, /*reuse_a=*/false, /*reuse_b=*/false);
}

// ---------------------------------------------------------------------------
// Kernel 1: O = X @ W^T, written head-major [B,H,S,AD], optionally scaled.
// Wave computes a 16(m) x 64(j) macro-tile: A fragment reused across 4 B tiles.
// A lane layout (16x4 f32): VGPR j, lane-half h -> K = j + 2h  => float2/lane.
// B (4x16) mirrors it: element (K, N=lane&15) = W[(j0+N)*D + k + K] => float2.
// ---------------------------------------------------------------------------
__global__ void __launch_bounds__(256) proj_head_major(
    const float* __restrict__ X,   // [4096, 1024]
    const float* __restrict__ W,   // [1024, 1024]
    float* __restrict__ O,         // [BH, S, AD]
    float scale)
{
  const int lane = threadIdx.x & 31;
  const int wave = threadIdx.x >> 5;
  const int tile = blockIdx.x * 8 + wave;      // 0..4095
  const int mt   = tile >> 4;                  // 0..255 : 16-row tile
  const int jt4  = tile & 15;                  // 0..15  : 64-col group == head
  const int m0 = mt * 16;
  const int j0 = jt4 * 64;
  const int r    = lane & 15;
  const int half = lane >> 4;

  const float* arow = X + (size_t)(m0 + r) * DMODEL + 2 * half;
  const float* b0 = W + (size_t)(j0 +  0 + r) * DMODEL + 2 * half;
  const float* b1 = W + (size_t)(j0 + 16 + r) * DMODEL + 2 * half;
  const float* b2 = W + (size_t)(j0 + 32 + r) * DMODEL + 2 * half;
  const float* b3 = W + (size_t)(j0 + 48 + r) * DMODEL + 2 * half;

  v8f acc0 = {}, acc1 = {}, acc2 = {}, acc3 = {};
  #pragma unroll 4
  for (int k = 0; k < DMODEL; k += 4) {
    v2f a  = *(const v2f*)(arow + k);
    v2f w0 = *(const v2f*)(b0 + k);
    v2f w1 = *(const v2f*)(b1 + k);
    v2f w2 = *(const v2f*)(b2 + k);
    v2f w3 = *(const v2f*)(b3 + k);
    acc0 = wmma_f32_k4(a, w0, acc0);
    acc1 = wmma_f32_k4(a, w1, acc1);
    acc2 = wmma_f32_k4(a, w2, acc2);
    acc3 = wmma_f32_k4(a, w3, acc3);
  }

  // C/D layout: VGPR i, lane-half h -> row m0+i+8h, col lane&15.
  const int h = jt4;                 // whole 64-col group is one head
  #pragma unroll
  for (int i = 0; i < 8; ++i) {
    const int n  = m0 + i + 8 * half;        // global row in [0, 4096)
    const int b_ = n >> 11;                  // batch = n / S
    const int s  = n & (SEQ - 1);
    float* dst = O + (((size_t)(b_ * NHEAD + h) * SEQ) + s) * ADIM;
    dst[ 0 + r] = acc0[i] * scale;
    dst[16 + r] = acc1[i] * scale;
    dst[32 + r] = acc2[i] * scale;
    dst[48 + r] = acc3[i] * scale;
  }
}

// ---------------------------------------------------------------------------
// Kernel 2: raw scores P[q,k] = Qh[q,:] . Kh[k,:]  (Q already has 1/8 scale).
// Wave computes 16(q) x 64(k) reusing the Q fragment across 4 K tiles.
// ---------------------------------------------------------------------------
__global__ void __launch_bounds__(256) attn_scores(
    const float* __restrict__ Qw,   // [BH, S, AD]
    const float* __restrict__ Kw,   // [BH, S, AD]
    float* __restrict__ P)          // [BH, S, S]
{
  const int lane = threadIdx.x & 31;
  const int wave = threadIdx.x >> 5;
  const int bh   = blockIdx.y;
  const int t    = blockIdx.x * 8 + wave;   // 0..4095
  const int qt   = t >> 5;                  // 0..127
  const int kt4  = t & 31;                  // 0..31 (64-wide k group)
  const int q0 = qt * 16;
  const int k0 = kt4 * 64;
  const int r    = lane & 15;
  const int half = lane >> 4;

  const float* qrow = Qw + ((size_t)bh * SEQ + q0 + r) * ADIM + 2 * half;
  const float* kr0 = Kw + ((size_t)bh * SEQ + k0 +  0 + r) * ADIM + 2 * half;
  const float* kr1 = Kw + ((size_t)bh * SEQ + k0 + 16 + r) * ADIM + 2 * half;
  const float* kr2 = Kw + ((size_t)bh * SEQ + k0 + 32 + r) * ADIM + 2 * half;
  const float* kr3 = Kw + ((size_t)bh * SEQ + k0 + 48 + r) * ADIM + 2 * half;

  v8f acc0 = {}, acc1 = {}, acc2 = {}, acc3 = {};
  #pragma unroll
  for (int a = 0; a < ADIM; a += 4) {
    v2f qa = *(const v2f*)(qrow + a);
    v2f x0 = *(const v2f*)(kr0 + a);
    v2f x1 = *(const v2f*)(kr1 + a);
    v2f x2 = *(const v2f*)(kr2 + a);
    v2f x3 = *(const v2f*)(kr3 + a);
    acc0 = wmma_f32_k4(qa, x0, acc0);
    acc1 = wmma_f32_k4(qa, x1, acc1);
    acc2 = wmma_f32_k4(qa, x2, acc2);
    acc3 = wmma_f32_k4(qa, x3, acc3);
  }

  float* pbase = P + (size_t)bh * SEQ * SEQ;
  #pragma unroll
  for (int i = 0; i < 8; ++i) {
    const int q = q0 + i + 8 * half;
    float* dst = pbase + (size_t)q * SEQ + k0;
    dst[ 0 + r] = acc0[i];
    dst[16 + r] = acc1[i];
    dst[32 + r] = acc2[i];
    dst[48 + r] = acc3[i];
  }
}

// ---------------------------------------------------------------------------
// Kernel 3: in-place row softmax over 2048 elements; one block per row.
// ---------------------------------------------------------------------------
__global__ void __launch_bounds__(256) row_softmax(float* __restrict__ P)
{
  __shared__ float red[256];
  float* p = P + (size_t)blockIdx.x * SEQ;
  const int t = threadIdx.x;

  float v[8];
  float m = -3.402823466e+38f;
  #pragma unroll
  for (int i = 0; i < 8; ++i) { v[i] = p[t + 256 * i]; m = fmaxf(m, v[i]); }

  red[t] = m; __syncthreads();
  for (int off = 128; off > 0; off >>= 1) {
    if (t < off) red[t] = fmaxf(red[t], red[t + off]);
    __syncthreads();
  }
  m = red[0]; __syncthreads();

  float sum = 0.0f;
  #pragma unroll
  for (int i = 0; i < 8; ++i) { v[i] = __expf(v[i] - m); sum += v[i]; }

  red[t] = sum; __syncthreads();
  for (int off = 128; off > 0; off >>= 1) {
    if (t < off) red[t] += red[t + off];
    __syncthreads();
  }
  const float inv = 1.0f / red[0];

  #pragma unroll
  for (int i = 0; i < 8; ++i) p[t + 256 * i] = v[i] * inv;
}

// ---------------------------------------------------------------------------
// Kernel 4: Z = P @ V per head. Wave covers 16(q) x 64(=full AD) with 4 accs.
// B fragment: element (K, N) = V[(k+K)*AD + a0+N] -> two per-lane b32 loads
// (contiguous across the 16 lanes of each half).
// ---------------------------------------------------------------------------
__global__ void __launch_bounds__(256) attn_pv(
    const float* __restrict__ P,    // [BH, S, S] probabilities
    const float* __restrict__ Vw,   // [BH, S, AD]
    float* __restrict__ Z)          // [BH, S, AD]
{
  const int lane = threadIdx.x & 31;
  const int wave = threadIdx.x >> 5;
  const int bh   = blockIdx.y;
  const int qt   = blockIdx.x * 8 + wave;   // 0..127
  const int q0 = qt * 16;
  const int r    = lane & 15;
  const int half = lane >> 4;

  const float* prow = P + ((size_t)bh * SEQ + q0 + r) * SEQ + 2 * half;
  const float* vb   = Vw + (size_t)bh * SEQ * ADIM + r;   // + col group + k*AD

  v8f acc0 = {}, acc1 = {}, acc2 = {}, acc3 = {};
  #pragma unroll 4
  for (int k = 0; k < SEQ; k += 4) {
    v2f a = *(const v2f*)(prow + k);
    const size_t row0 = (size_t)(k + 2 * half) * ADIM;
    const size_t row1 = (size_t)(k + 1 + 2 * half) * ADIM;
    v2f b0, b1, b2, b3;
    b0.x = vb[row0 +  0]; b0.y = vb[row1 +  0];
    b1.x = vb[row0 + 16]; b1.y = vb[row1 + 16];
    b2.x = vb[row0 + 32]; b2.y = vb[row1 + 32];
    b3.x = vb[row0 + 48]; b3.y = vb[row1 + 48];
    acc0 = wmma_f32_k4(a, b0, acc0);
    acc1 = wmma_f32_k4(a, b1, acc1);
    acc2 = wmma_f32_k4(a, b2, acc2);
    acc3 = wmma_f32_k4(a, b3, acc3);
  }

  float* zbase = Z + (size_t)bh * SEQ * ADIM;
  #pragma unroll
  for (int i = 0; i < 8; ++i) {
    const int q = q0 + i + 8 * half;
    float* dst = zbase + (size_t)q * ADIM;
    dst[ 0 + r] = acc0[i];
    dst[16 + r] = acc1[i];
    dst[32 + r] = acc2[i];
    dst[48 + r] = acc3[i];
  }
}

// ---------------------------------------------------------------------------
extern "C" void kernel_launch(void* const* d_in, const int* in_sizes, int n_in,
                              void* d_out, int out_size, void* d_ws, size_t ws_size,
                              hipStream_t stream) {
  const float* inQ = (const float*)d_in[0];
  const float* inK = (const float*)d_in[1];
  const float* inV = (const float*)d_in[2];
  const float* Wq  = (const float*)d_in[3];
  const float* Wk  = (const float*)d_in[4];
  const float* Wv  = (const float*)d_in[5];

  float* out = (float*)d_out;
  float* Z   = out;                 // [B,H,S,AD]  : 4,194,304 floats
  float* P   = out + ZSIZE;         // [B,H,S,S]   : 134,217,728 floats

  float* ws = (float*)d_ws;         // 3 x 16.78 MB projected tensors
  float* Qw = ws;
  float* Kw = ws + (size_t)ZSIZE;
  float* Vw = ws + (size_t)2 * ZSIZE;

  const dim3 blk(256);

  // QKV projections (Q pre-scaled by 1/sqrt(64))
  proj_head_major<<<dim3(512), blk, 0, stream>>>(inQ, Wq, Qw, 0.125f);
  proj_head_major<<<dim3(512), blk, 0, stream>>>(inK, Wk, Kw, 1.0f);
  proj_head_major<<<dim3(512), blk, 0, stream>>>(inV, Wv, Vw, 1.0f);

  // Raw scores -> softmax in place (d_out score region) -> Z
  attn_scores<<<dim3(512, BH), blk, 0, stream>>>(Qw, Kw, P);
  row_softmax<<<dim3(BH * SEQ), blk, 0, stream>>>(P);
  attn_pv<<<dim3(16, BH), blk, 0, stream>>>(P, Vw, Z);
}